// Wavenet_54597624267053
// MI455X (gfx1250) — compile-verified
//
#include <hip/hip_runtime.h>
#include <hip/hip_bf16.h>
#include <math.h>

// Problem constants (fixed by setup_inputs in the reference)
#define NLAYERS 6
#define BATCH   8
#define LLEN    64000
#define OUTC    256

// ---- Kernel 1: full 6-layer residual chain, tiled in LDS with 42-halo ----
#define CHUNK 2048
#define HALO  42                 // sum of (2d+1) over d = 1,2,4,8,1,2
#define EXT   (CHUNK + HALO)

__global__ __launch_bounds__(256)
void wavenet_chain_kernel(const float* __restrict__ X,
                          const float* __restrict__ w1s, const float* __restrict__ b1s,
                          const float* __restrict__ w2s, const float* __restrict__ b2s,
                          const float* __restrict__ w3s, const float* __restrict__ b3s,
                          float* __restrict__ Xf)
{
    __shared__ float bufA[EXT];
    __shared__ float bufB[EXT];

    const int nchunks = (LLEN + CHUNK - 1) / CHUNK;   // 32
    const int b     = blockIdx.x / nchunks;
    const int start = (blockIdx.x % nchunks) * CHUNK;
    const int tid   = threadIdx.x;

    // Load [start-HALO, start+CHUNK); out-of-range -> 0 (matches conv zero pad)
    for (int t = tid; t < EXT; t += 256) {
        int g = start - HALO + t;
        bufA[t] = (g >= 0 && g < LLEN) ? X[b * LLEN + g] : 0.0f;
    }
    __syncthreads();

    float* src = bufA;
    float* dst = bufB;
    int off = 0;
    for (int i = 0; i < NLAYERS; ++i) {
        const int   d   = 1 << (i & 3);                 // 1,2,4,8,1,2
        const float w10 = w1s[i*3+0], w11 = w1s[i*3+1], w12 = w1s[i*3+2];
        const float w20 = w2s[i*3+0], w21 = w2s[i*3+1], w22 = w2s[i*3+2];
        const float b1  = b1s[i], b2 = b2s[i], w3 = w3s[i], b3 = b3s[i];
        off += 2*d + 1;                                  // valid region shrinks
        for (int t = off + tid; t < EXT; t += 256) {
            // conv taps look back at j-1, j-(d+1), j-(2d+1)
            float xm1  = src[t - 1];
            float xmd  = src[t - d - 1];
            float xm2d = src[t - 2*d - 1];
            float c1 = fmaf(w10, xm2d, fmaf(w11, xmd, fmaf(w12, xm1, b1)));
            float c2 = fmaf(w20, xm2d, fmaf(w21, xmd, fmaf(w22, xm1, b2)));
            float tg = tanhf(c1);
            float sg = 1.0f / (1.0f + __expf(-c2));
            float y  = src[t] + fmaf(tg * sg, w3, b3);
            int g = start - HALO + t;
            dst[t] = (g < 0) ? 0.0f : y;   // re-zero left pad for next layer's conv
        }
        __syncthreads();
        float* tmp = src; src = dst; dst = tmp;
    }
    // off == HALO here; write valid chunk
    for (int t = HALO + tid; t < EXT; t += 256) {
        int g = start - HALO + t;
        if (g < LLEN) Xf[b * LLEN + g] = src[t];
    }
}

// ---- Kernel 2: logits via V_WMMA_F32_16X16X4_F32 + stable softmax, NT stores ----
typedef __attribute__((ext_vector_type(2))) float v2f;
typedef __attribute__((ext_vector_type(8))) float v8f;

#define SEGS_PER_B (LLEN / 32)   // 2000, exact

__global__ __launch_bounds__(256)
void wavenet_softmax_kernel(const float* __restrict__ Xf,
                            const float* __restrict__ wout,
                            const float* __restrict__ bout,
                            float* __restrict__ out)
{
    __shared__ float sw[OUTC];
    __shared__ float sb[OUTC];
    __shared__ float smx[8][32];
    __shared__ float srv[8][32];
    __shared__ float sx [8][32];
    __shared__ float str[8][16 * 36];   // per-wave 16c x 32l transpose tile, padded stride

    const int tid = threadIdx.x;
    for (int t = tid; t < OUTC; t += 256) { sw[t] = wout[t]; sb[t] = bout[t]; }

    const int wave  = tid >> 5;
    const int lane  = tid & 31;
    const int seg   = blockIdx.x * 8 + wave;          // grid sized exactly: no guard
    const int bIdx  = seg / SEGS_PER_B;
    const int lbase = (seg % SEGS_PER_B) * 32;

    __syncthreads();

    // Phase A: per-l max and inverse denominator (one lane = one l)
    const float x = Xf[bIdx * LLEN + lbase + lane];
    float mx = -3.402823466e38f;
    for (int c = 0; c < OUTC; ++c) mx = fmaxf(mx, fmaf(x, sw[c], sb[c]));
    float s = 0.0f;
    for (int c = 0; c < OUTC; ++c) s += __expf(fmaf(x, sw[c], sb[c]) - mx);
    smx[wave][lane] = mx;
    srv[wave][lane] = 1.0f / s;
    sx [wave][lane] = x;
    __syncthreads();

    // Per-lane D-layout metadata: D vgpr r, lanes 0-15 -> M=r, lanes 16-31 -> M=r+8
    const int hi = lane >> 4;
    const int lo = lane & 15;
    float mx0[8], rv0[8], mx1[8], rv1[8];
    for (int r = 0; r < 8; ++r) {
        mx0[r] = smx[wave][r + 8*hi];        rv0[r] = srv[wave][r + 8*hi];
        mx1[r] = smx[wave][16 + r + 8*hi];   rv1[r] = srv[wave][16 + r + 8*hi];
    }
    // A 16x4 f32 layout: lanes 0-15 hold M=lane, vgpr0=K0, vgpr1=K1; lanes 16-31 K2,K3.
    // Rank-1: only K=0 column is nonzero (= x), rest 0.
    const float xlo = (lane < 16) ? x : 0.0f;
    const float xhi = (lane < 16) ? sx[wave][16 + lo] : 0.0f;
    v2f A0; A0[0] = xlo; A0[1] = 0.0f;      // l = lbase + 0..15
    v2f A1; A1[0] = xhi; A1[1] = 0.0f;      // l = lbase + 16..31

    for (int g = 0; g < 16; ++g) {          // 16 channel groups of 16
        // B 4x16: only K=0 row nonzero (= wout[group])
        float wv = (lane < 16) ? sw[g*16 + lane] : 0.0f;
        v2f Bm; Bm[0] = wv; Bm[1] = 0.0f;
        const float bb = sb[g*16 + lo];
        v8f C0, C1;
        for (int r = 0; r < 8; ++r) { C0[r] = bb - mx0[r]; C1[r] = bb - mx1[r]; }

        // D = x (x) wout + (bout - rowmax), full f32
        v8f D0 = __builtin_amdgcn_wmma_f32_16x16x4_f32(
            false, A0, false, Bm, (short)0, C0, false, false);
        v8f D1 = __builtin_amdgcn_wmma_f32_16x16x4_f32(
            false, A1, false, Bm, (short)0, C1, false, false);

        // exp, scale by 1/denominator, transpose via LDS (c-major rows, 32l cols)
        for (int r = 0; r < 8; ++r) {
            str[wave][lo*36 + (r + 8*hi)]        = __expf(D0[r]) * rv0[r];
            str[wave][lo*36 + 16 + (r + 8*hi)]   = __expf(D1[r]) * rv1[r];
        }
        __syncthreads();

        // Coalesced 128B-per-wave non-temporal stores: lane = l offset
        for (int cc = 0; cc < 16; ++cc) {
            float v = str[wave][cc*36 + lane];
            size_t idx = (size_t)(bIdx * OUTC + g*16 + cc) * LLEN + lbase + lane;
            __builtin_nontemporal_store(v, &out[idx]);
        }
        __syncthreads();
    }
}

extern "C" void kernel_launch(void* const* d_in, const int* in_sizes, int n_in,
                              void* d_out, int out_size, void* d_ws, size_t ws_size,
                              hipStream_t stream)
{
    const float* X    = (const float*)d_in[0];
    const float* w1s  = (const float*)d_in[1];
    const float* b1s  = (const float*)d_in[2];
    const float* w2s  = (const float*)d_in[3];
    const float* b2s  = (const float*)d_in[4];
    const float* w3s  = (const float*)d_in[5];
    const float* b3s  = (const float*)d_in[6];
    const float* wout = (const float*)d_in[7];
    const float* bout = (const float*)d_in[8];
    float* out = (float*)d_out;
    float* Xf  = (float*)d_ws;   // BATCH*LLEN floats = 2 MB scratch

    const int nchunks = (LLEN + CHUNK - 1) / CHUNK;          // 32
    wavenet_chain_kernel<<<BATCH * nchunks, 256, 0, stream>>>(
        X, w1s, b1s, w2s, b2s, w3s, b3s, Xf);

    const int nblocks = BATCH * SEGS_PER_B / 8;              // 2000
    wavenet_softmax_kernel<<<nblocks, 256, 0, stream>>>(Xf, wout, bout, out);
}